// SplineFilter_76390288327115
// MI455X (gfx1250) — compile-verified
//
#include <hip/hip_runtime.h>
#include <hip/hip_bf16.h>

typedef float v2f __attribute__((ext_vector_type(2)));
typedef float v8f __attribute__((ext_vector_type(8)));

namespace {
constexpr int FCH  = 128;            // feature channels
constexpr float KH = 2.0f / 15.0f;   // uniform knot spacing over [0,2]
}

// Branch-free Cox-de Boor for uniform knots t_i = i*KH, i=0..15, degree 3.
// Uses (x - i*KH)/(d*KH) = x/(d*KH) - i so each recursion term is sub/sub+mul+fma.
__device__ __forceinline__ void bspline_basis12(float x, float b[15]) {
    #pragma unroll
    for (int i = 0; i < 15; ++i) {
        const float t0 = (float)i * KH;
        const float t1 = (float)(i + 1) * KH;
        b[i] = (x >= t0 && x < t1) ? 1.0f : 0.0f;
    }
    #pragma unroll
    for (int d = 1; d <= 3; ++d) {
        const float xi = x * (15.0f / (2.0f * (float)d));   // x / (d*KH)
        #pragma unroll
        for (int i = 0; i < 15 - d; ++i) {
            const float left  = xi - (float)i;
            const float right = (float)(i + d + 1) - xi;
            b[i] = __builtin_fmaf(left, b[i], right * b[i + 1]);
        }
    }
}

// One wave32 handles a chunk of 16 rows x 128 channels.
//   spectral[16x16 tile] = A(16x12 basis) x B(12x16 coeffs)  via 3x WMMA_F32_16X16X4_F32
//   out = eval_x * spectral^2
__global__ void __launch_bounds__(256)
spline_filter_wmma_kernel(const float* __restrict__ eval_x,
                          const float* __restrict__ eval_eigs,
                          const float* __restrict__ coeffs,
                          float* __restrict__ out,
                          int total_rows) {
    const int lane          = threadIdx.x & 31;
    const int waveInBlock   = threadIdx.x >> 5;
    const int wavesPerBlock = blockDim.x >> 5;
    const int globalWave    = blockIdx.x * wavesPerBlock + waveInBlock;
    const int numWaves      = gridDim.x * wavesPerBlock;

    const int  lo   = lane & 15;       // N index within 16-wide tile / row-within-chunk
    const int  hi   = lane >> 4;       // half-wave select
    const bool hib  = (hi != 0);
    const int  kOff = hi * 2;          // K offset per ISA A/B fragment layout

    // ---- B fragments: coeffs[12][128] -> 8 N-tiles x 3 K-blocks, kept in VGPRs ----
    // Layout (4x16 fp32 B): VGPR0 = {K=k0 | lanes 0-15, K=k0+2 | lanes 16-31},
    //                       VGPR1 = {K=k0+1, K=k0+3}.
    v2f Bf[8][3];
    #pragma unroll
    for (int t = 0; t < 8; ++t) {
        #pragma unroll
        for (int j = 0; j < 3; ++j) {
            const int k0 = 4 * j + kOff;
            Bf[t][j].x = coeffs[(k0 + 0) * FCH + t * 16 + lo];
            Bf[t][j].y = coeffs[(k0 + 1) * FCH + t * 16 + lo];
        }
    }

    const int totalChunks = total_rows >> 4;  // 16 rows per chunk

    // Wave-uniform grid-stride loop (EXEC stays all-ones for WMMA).
    for (int chunk = globalWave; chunk < totalChunks; chunk += numWaves) {
        const int rowBase = chunk << 4;

        // Each lane evaluates basis for row (rowBase + lo); both half-waves
        // duplicate the rows but pick different K slices for the A fragment.
        const float xe = eval_eigs[rowBase + lo];
        float bs[15];
        bspline_basis12(xe, bs);

        // A fragment (16x4 fp32): lanes 0-15 carry K = 4j,4j+1; lanes 16-31 carry
        // K = 4j+2,4j+3. Constant-index 2-way selects only (no dynamic indexing).
        v2f Af[3];
        Af[0].x = hib ? bs[2]  : bs[0];
        Af[0].y = hib ? bs[3]  : bs[1];
        Af[1].x = hib ? bs[6]  : bs[4];
        Af[1].y = hib ? bs[7]  : bs[5];
        Af[2].x = hib ? bs[10] : bs[8];
        Af[2].y = hib ? bs[11] : bs[9];

        const float* __restrict__ xrow = eval_x + (size_t)rowBase * FCH;
        float* __restrict__ orow       = out    + (size_t)rowBase * FCH;

        #pragma unroll
        for (int t = 0; t < 8; ++t) {
            v8f c = {};
            // D = A x B + C ; accumulate K = 0..3, 4..7, 8..11
            c = __builtin_amdgcn_wmma_f32_16x16x4_f32(
                    false, Af[0], false, Bf[t][0], (short)0, c, false, false);
            c = __builtin_amdgcn_wmma_f32_16x16x4_f32(
                    false, Af[1], false, Bf[t][1], (short)0, c, false, false);
            c = __builtin_amdgcn_wmma_f32_16x16x4_f32(
                    false, Af[2], false, Bf[t][2], (short)0, c, false, false);

            // C/D layout: lane l, reg v -> (M = v + 8*hi, N = lo)
            #pragma unroll
            for (int v = 0; v < 8; ++v) {
                const int r = v + hi * 8;
                const size_t idx = (size_t)r * FCH + t * 16 + lo;
                const float xv = xrow[idx];
                const float f  = c[v];
                orow[idx] = xv * f * f;
            }
        }
    }
}

extern "C" void kernel_launch(void* const* d_in, const int* in_sizes, int n_in,
                              void* d_out, int out_size, void* d_ws, size_t ws_size,
                              hipStream_t stream) {
    (void)n_in; (void)d_ws; (void)ws_size; (void)out_size;
    const float* eval_x    = (const float*)d_in[0];  // [B, N, 128] f32
    const float* eval_eigs = (const float*)d_in[1];  // [B, N]      f32
    const float* coeffs    = (const float*)d_in[2];  // [12, 128]   f32
    float* out             = (float*)d_out;          // [B, N, 128] f32

    const int total_rows = in_sizes[1];              // B*N = 262144
    const int threads = 256;                         // 8 waves/block
    const int wavesPerBlock = threads / 32;
    const int chunks = total_rows >> 4;
    int blocks = (chunks + wavesPerBlock - 1) / wavesPerBlock;
    if (blocks > 1024) blocks = 1024;                // persistent waves amortize B-frag loads
    if (blocks < 1) blocks = 1;

    spline_filter_wmma_kernel<<<blocks, threads, 0, stream>>>(
        eval_x, eval_eigs, coeffs, out, total_rows);
}